// QLayer_16707422781763
// MI455X (gfx1250) — compile-verified
//
#include <hip/hip_runtime.h>
#include <math.h>

// CDNA5 / gfx1250. wave32. WMMA f32 16x16x4 (exact fp32 matmul).
typedef __attribute__((ext_vector_type(2))) float v2f;
typedef __attribute__((ext_vector_type(8))) float v8f;

#define NQ 8
#define LSTRIDE 17   // 16 + 1 pad: kills LDS bank conflicts on row-indexed loads

// Read-index map for the CNOT ring applied to the state:
//   s_after[y] = s_before[R(y)],  R = f1∘f2∘...∘f8 (each CNOT is an involution),
// ring order in the reference: CNOT(0,1),(1,2),...,(6,7),(7,0); qubit q <-> bit(7-q).
__device__ __forceinline__ int cnot_read_idx(int y) {
  y ^= (y & 0x01) << 7;  // CNOT(7,0): if b0 flip b7   (applied innermost = last gate)
  y ^= (y & 0x02) >> 1;  // CNOT(6,7): if b1 flip b0
  y ^= (y & 0x04) >> 1;  // CNOT(5,6)
  y ^= (y & 0x08) >> 1;  // CNOT(4,5)
  y ^= (y & 0x10) >> 1;  // CNOT(3,4)
  y ^= (y & 0x20) >> 1;  // CNOT(2,3)
  y ^= (y & 0x40) >> 1;  // CNOT(1,2)
  y ^= (y & 0x80) >> 1;  // CNOT(0,1): if b7 flip b6   (first gate, outermost)
  return y;
}

// One wave simulates the whole circuit (output is token-independent: all RX gates
// act on |+> eigenstates -> global phase only; state stays real).
__global__ __launch_bounds__(32) void qsim_one_token(const float* __restrict__ w,
                                                     float* __restrict__ z_out) {
  __shared__ float S[16 * LSTRIDE];   // real state, 16x16
  __shared__ float KR[16 * LSTRIDE];  // kron(RY q0..q3) for current layer
  __shared__ float KC[16 * LSTRIDE];  // kron(RY q4..q7)

  const int t = threadIdx.x;  // 0..31
  const int m = t & 15;       // M (A-operand) / N (B,D operand) index
  const int h = t >> 4;       // lane half

  // ---- initial state after encode + RX layer + first RY layer ----
  // RX(x)|+> and RX(w)|+> are global phases -> per-qubit state is real:
  // u_q = RY(w[8+q]) * (1,1)/sqrt2 = ((c-s), (c+s))/sqrt2
  float g0[NQ], g1[NQ];
  for (int q = 0; q < NQ; ++q) {
    float th = 0.5f * w[8 + q];
    float cb = __cosf(th), sb = __sinf(th);
    g0[q] = (cb - sb) * 0.70710678118654752f;
    g1[q] = (cb + sb) * 0.70710678118654752f;
  }
  // amp(i) = prod_q u_q[bit(7-q) of i];  S[r][c], r = i>>4, c = i&15
  for (int e = 0; e < 8; ++e) {
    int i = t * 8 + e;
    int r = i >> 4, c = i & 15;
    float a = 1.0f;
    for (int q = 0; q < 4; ++q) a *= ((r >> (3 - q)) & 1) ? g1[q] : g0[q];
    for (int q = 4; q < 8; ++q) a *= ((c >> (7 - q)) & 1) ? g1[q] : g0[q];
    S[r * LSTRIDE + c] = a;
  }

  // Precompute CNOT-permuted LDS addresses for the row-matmul B operand.
  // B block j, elem e: B[4j+2h+e][m] = S_old[R((k<<4)|m)]
  int baddr[8];
  for (int j = 0; j < 4; ++j)
    for (int e = 0; e < 2; ++e) {
      int k = 4 * j + 2 * h + e;
      int src = cnot_read_idx((k << 4) | m);
      baddr[j * 2 + e] = (src >> 4) * LSTRIDE + (src & 15);
    }
  __syncthreads();

  v8f acc = {};
  for (int l = 2; l < 8; ++l) {
    // ---- build kron(RY) operators for this layer ----
    float cr[4], sr[4], cc[4], sc[4];
    for (int q = 0; q < 4; ++q) {
      float a = 0.5f * w[8 * l + q];
      cr[q] = __cosf(a); sr[q] = __sinf(a);
      float b = 0.5f * w[8 * l + 4 + q];
      cc[q] = __cosf(b); sc[q] = __sinf(b);
    }
    for (int e = 0; e < 8; ++e) {
      int i = t * 8 + e;
      int r = i >> 4, k = i & 15;
      float ar = 1.0f, ac = 1.0f;
      for (int q = 0; q < 4; ++q) {
        int rb = (r >> (3 - q)) & 1, kb = (k >> (3 - q)) & 1;
        // RY = [[c,-s],[s,c]]
        ar *= (rb == kb) ? cr[q] : (rb ? sr[q] : -sr[q]);
        ac *= (rb == kb) ? cc[q] : (rb ? sc[q] : -sc[q]);
      }
      KR[r * LSTRIDE + k] = ar;
      KC[r * LSTRIDE + k] = ac;
    }
    __syncthreads();

    // ---- row matmul: S1 = KR x (CNOT-ring-permuted S) ----
    // A lane layout (16x4 f32): {A[m][4j+2h], A[m][4j+2h+1]}; B mirrored; 4 chained WMMAs.
    acc = (v8f){};
    for (int j = 0; j < 4; ++j) {
      int k0 = 4 * j + 2 * h;
      v2f a, b;
      a.x = KR[m * LSTRIDE + k0];
      a.y = KR[m * LSTRIDE + k0 + 1];
      b.x = S[baddr[2 * j]];
      b.y = S[baddr[2 * j + 1]];
      acc = __builtin_amdgcn_wmma_f32_16x16x4_f32(false, a, false, b,
                                                  (short)0, acc, false, false);
    }
    __syncthreads();
    // D layout -> plain LDS layout: lane t holds rows {v+8h}, col m
    for (int v = 0; v < 8; ++v) S[(v + 8 * h) * LSTRIDE + m] = acc[v];
    __syncthreads();

    // ---- col matmul: S2 = S1 x KC^T  (B[k][n] = KC[n][k]) ----
    v8f acc2 = {};
    for (int j = 0; j < 4; ++j) {
      int k0 = 4 * j + 2 * h;
      v2f a, b;
      a.x = S[m * LSTRIDE + k0];
      a.y = S[m * LSTRIDE + k0 + 1];
      b.x = KC[m * LSTRIDE + k0];   // n == m for B lanes
      b.y = KC[m * LSTRIDE + k0 + 1];
      acc2 = __builtin_amdgcn_wmma_f32_16x16x4_f32(false, a, false, b,
                                                   (short)0, acc2, false, false);
    }
    __syncthreads();
    for (int v = 0; v < 8; ++v) S[(v + 8 * h) * LSTRIDE + m] = acc2[v];
    acc = acc2;
    __syncthreads();
  }

  // ---- <Z_q> straight from the final accumulators ----
  // lane t, elem v: amplitude of i = ((v+8h)<<4) | m
  float P = 0.f, z1 = 0.f, z2 = 0.f, z3 = 0.f;
  for (int v = 0; v < 8; ++v) {
    float p = acc[v] * acc[v];
    P += p;
    z1 += ((v >> 2) & 1) ? -p : p;  // qubit1 <-> bit6 <-> v bit2
    z2 += ((v >> 1) & 1) ? -p : p;  // qubit2
    z3 += ((v     ) & 1) ? -p : p;  // qubit3
  }
  float z[8];
  z[0] = h ? -P : P;                 // qubit0 <-> bit7 <-> half
  z[1] = z1; z[2] = z2; z[3] = z3;
  z[4] = ((m >> 3) & 1) ? -P : P;    // qubit4 <-> bit3 of col
  z[5] = ((m >> 2) & 1) ? -P : P;
  z[6] = ((m >> 1) & 1) ? -P : P;
  z[7] = ((m     ) & 1) ? -P : P;
  for (int q = 0; q < 8; ++q)
    for (int off = 16; off > 0; off >>= 1)
      z[q] += __shfl_xor(z[q], off, 32);
  if (t == 0)
    for (int q = 0; q < 8; ++q) z_out[q] = z[q];
}

// Broadcast the 8-float row to all tokens: pure streaming stores (HBM-rate fill).
__global__ __launch_bounds__(256) void bcast_rows(const float4* __restrict__ z,
                                                  float4* __restrict__ out, int ntok) {
  float4 z0 = z[0];
  float4 z1 = z[1];
  int tkn = blockIdx.x * blockDim.x + threadIdx.x;
  if (tkn < ntok) {
    out[2 * tkn]     = z0;
    out[2 * tkn + 1] = z1;
  }
}

extern "C" void kernel_launch(void* const* d_in, const int* in_sizes, int n_in,
                              void* d_out, int out_size, void* d_ws, size_t ws_size,
                              hipStream_t stream) {
  // d_in[0] = x (unused: provably a global phase), d_in[1] = w (64 floats)
  const float* w = (const float*)d_in[1];
  float* z = (float*)d_ws;  // 8 floats of scratch

  qsim_one_token<<<1, 32, 0, stream>>>(w, z);

  int ntok = out_size / NQ;
  bcast_rows<<<(ntok + 255) / 256, 256, 0, stream>>>((const float4*)z,
                                                     (float4*)d_out, ntok);
}